// CIF_8847632630315
// MI455X (gfx1250) — compile-verified
//
#include <hip/hip_runtime.h>
#include <math.h>

// Problem constants (match reference)
#define BB     16
#define TT     2048
#define DD     256
#define WW     5
#define YMAXX  256
#define KK     (DD * WW)   // 1280
#define PADW   2
#define EPSF   1e-12f

// d_out layout (floats): cv [B,YMAX,D] | aws [B,YMAX,T] | alpha [B,T]
#define CV_OFF     0
#define AWS_OFF    (BB * YMAXX * DD)               // 1,048,576
#define ALPHA_OFF  (AWS_OFF + BB * YMAXX * TT)     // 9,437,184

typedef __attribute__((ext_vector_type(2))) float v2f;
typedef __attribute__((ext_vector_type(8))) float v8f;

// ---------------------------------------------------------------------------
// Kernel 0: pack conv_w [Dout=256][Din=256][W=5] into WMMA-fragment order.
// Logical GEMM B is B[kk][n] = conv_w[n][k][w], kk = w*256 + k.
// Fragment layout (float2 per lane): for K-quad k4 = kk/4, half h, lane col n:
//   Bfrag[(k4*2 + h)*256 + n] = { B[4k4+2h][n], B[4k4+2h+1][n] }
// so a lane's two K-values for one 16x16 tile are one global_load_b64, and a
// wave's 16 lanes read 128 contiguous bytes.
// ---------------------------------------------------------------------------
__global__ __launch_bounds__(DD)
void cif_pack_w(const float* __restrict__ conv_w, float* __restrict__ Bfrag) {
    const int kk = blockIdx.x;      // 0..1279
    const int n  = threadIdx.x;     // 0..255
    const int w  = kk / DD;
    const int k  = kk % DD;
    const int p  = kk & 1;          // position within the float2
    const int h  = (kk >> 1) & 1;   // half (lanes 0-15 vs 16-31)
    const int k4 = kk >> 2;         // K-quad index
    Bfrag[(((size_t)(k4 * 2 + h)) * DD + n) * 2 + p] =
        conv_w[(size_t)n * (DD * WW) + k * WW + w];
}

// ---------------------------------------------------------------------------
// Kernel 1: conv-as-GEMM (fp32 WMMA 16x16x4) + bias + LayerNorm + ReLU +
//           proj + sigmoid + elens mask  ->  alpha [B,T]
// One block = one batch b, 16 time rows. 256 threads = 8 waves.
// Wave w computes output columns [32w, 32w+32) as two 16x16 WMMA tiles.
// ---------------------------------------------------------------------------
#define XPITCH 257   // 257 % 64 == 1 -> conflict-free column walks
__global__ __launch_bounds__(256)
void cif_alpha_kernel(const float* __restrict__ eouts,
                      const float* __restrict__ Bfrag,
                      const float* __restrict__ conv_b,
                      const float* __restrict__ ln_g,
                      const float* __restrict__ ln_b,
                      const float* __restrict__ proj_w,
                      const float* __restrict__ proj_b,
                      const int*   __restrict__ elens,
                      float*       __restrict__ alphaOut) {
    __shared__ float xtile[20 * XPITCH];   // rows t0-2 .. t0+17, zero padded
    __shared__ float ytile[16 * XPITCH];   // 16 x 256 conv outputs
    __shared__ float red0[256], red1[256];
    __shared__ float s_mu[16], s_rs[16];

    const int tid = threadIdx.x;
    const int b   = blockIdx.y;
    const int t0  = blockIdx.x * 16;

    // ---- stage x tile into LDS (coalesced, zero-padded at sequence edges)
    for (int idx = tid; idx < 20 * DD; idx += 256) {
        const int r = idx >> 8, c = idx & 255;
        const int t = t0 + r - PADW;
        float v = 0.0f;
        if (t >= 0 && t < TT) v = eouts[((size_t)b * TT + t) * DD + c];
        xtile[r * XPITCH + c] = v;
    }
    __syncthreads();

    // ---- GEMM: [16 x 1280] * [1280 x 256], K in steps of 4 (f32 WMMA)
    const int wave = tid >> 5;
    const int lane = tid & 31;
    const int l    = lane & 15;
    const int half = lane >> 4;
    const int n0   = wave * 32;

    v8f acc0 = {0.f,0.f,0.f,0.f,0.f,0.f,0.f,0.f};
    v8f acc1 = {0.f,0.f,0.f,0.f,0.f,0.f,0.f,0.f};

    // per-lane base into the fragment-packed B: K-quad 0, this half, col n0+l
    const float2* bbase = (const float2*)Bfrag + (size_t)half * DD + n0 + l;

    for (int w = 0; w < WW; ++w) {
        // A row for this lane: A[m][w*256+k] == xtile[m + w][k], m = l
        const float* ar = &xtile[(l + w) * XPITCH];
        for (int k = 0; k < DD; k += 4) {
            v2f a;
            a[0] = ar[k + 2 * half];          // K = 0 / 2
            a[1] = ar[k + 2 * half + 1];      // K = 1 / 3
            const int k4 = (w * DD + k) >> 2;
            const float2* bp = bbase + (size_t)k4 * 2 * DD;
            const float2 B0 = bp[0];          // tile 0: N = n0 + l      (b64)
            const float2 B1 = bp[16];         // tile 1: N = n0 + 16 + l (b64)
            v2f b0, b1;
            b0[0] = B0.x; b0[1] = B0.y;
            b1[0] = B1.x; b1[1] = B1.y;
            acc0 = __builtin_amdgcn_wmma_f32_16x16x4_f32(
                false, a, false, b0, (short)0, acc0, false, false);
            acc1 = __builtin_amdgcn_wmma_f32_16x16x4_f32(
                false, a, false, b1, (short)0, acc1, false, false);
        }
    }

    // ---- spill C tiles to LDS: VGPR v -> row (v + 8*half), col n0(+16) + l
    #pragma unroll
    for (int v = 0; v < 8; ++v) {
        const int m = v + 8 * half;
        ytile[m * XPITCH + n0 + l]      = acc0[v];
        ytile[m * XPITCH + n0 + 16 + l] = acc1[v];
    }
    __syncthreads();

    // ---- fused bias + LayerNorm + ReLU + projection + sigmoid
    const int m   = tid >> 4;   // row 0..15
    const int sub = tid & 15;   // 16 threads per row
    float sum = 0.0f, sq = 0.0f;
    for (int n = sub; n < DD; n += 16) {
        const float y = ytile[m * XPITCH + n] + conv_b[n];
        sum += y; sq += y * y;
    }
    red0[tid] = sum; red1[tid] = sq;
    __syncthreads();
    if (sub == 0) {
        float S = 0.0f, Q = 0.0f;
        for (int j = 0; j < 16; ++j) { S += red0[m * 16 + j]; Q += red1[m * 16 + j]; }
        const float mu  = S * (1.0f / DD);
        const float var = Q * (1.0f / DD) - mu * mu;
        s_mu[m] = mu;
        s_rs[m] = rsqrtf(var + EPSF);
    }
    __syncthreads();
    const float mu = s_mu[m], rs = s_rs[m];
    float dot = 0.0f;
    for (int n = sub; n < DD; n += 16) {
        const float y = ytile[m * XPITCH + n] + conv_b[n];
        float v = (y - mu) * rs * ln_g[n] + ln_b[n];
        v = fmaxf(v, 0.0f);
        dot += v * proj_w[n];
    }
    red0[tid] = dot;
    __syncthreads();
    if (sub == 0) {
        float Z = 0.0f;
        for (int j = 0; j < 16; ++j) Z += red0[m * 16 + j];
        float a = 1.0f / (1.0f + __expf(-(Z + proj_b[0])));
        const int t = t0 + m;
        if (t >= elens[b]) a = 0.0f;
        alphaOut[(size_t)b * TT + t] = a;
    }
}

// ---------------------------------------------------------------------------
// Kernel 2: CIF scan. Single block, 512 threads = 16 waves (wave b = batch b).
// Lane holds 8 contiguous D-values of the integrate state in registers.
// Cross-batch "fast" flag reduced through LDS each step (matches reference).
// ---------------------------------------------------------------------------
__global__ __launch_bounds__(512)
void cif_scan_kernel(const float* __restrict__ eouts,
                     const float* __restrict__ alpha,
                     const int*   __restrict__ elens,
                     const int*   __restrict__ ylens,
                     float*       __restrict__ cv,
                     float*       __restrict__ aws) {
    __shared__ float s_red[512];
    __shared__ float s_scale[BB];
    __shared__ float s_accum[BB];
    __shared__ int   s_ntok[BB];
    __shared__ int   s_flag[BB];
    __shared__ int   s_fast;

    const int tid  = threadIdx.x;
    const int b    = tid >> 5;
    const int lane = tid & 31;
    const int el   = elens[b];
    const int yl   = ylens[b];

    // ---- alpha normalization: scale = ylens[b] / sum_t alpha[b,t]
    float sum = 0.0f;
    for (int t = lane; t < TT; t += 32) sum += alpha[b * TT + t];
    s_red[tid] = sum;
    __syncthreads();
    if (lane == 0) {
        float S = 0.0f;
        for (int j = 0; j < 32; ++j) S += s_red[(b << 5) + j];
        s_scale[b] = (float)yl / S;
        s_accum[b] = 0.0f;
        s_ntok[b]  = 0;
    }
    __syncthreads();

    float st[8];
    #pragma unroll
    for (int i = 0; i < 8; ++i) st[i] = 0.0f;
    const int d0 = lane * 8;

    for (int t = 0; t < TT; ++t) {
        const float accum = s_accum[b];
        const int   n_tok = s_ntok[b];
        const float alpha_j   = alpha[b * TT + t] * s_scale[b];
        const float accum_new = accum + alpha_j;

        if (lane == 0) s_flag[b] = (accum_new >= 1.0f) ? 1 : 0;
        __syncthreads();
        if (tid == 0) {
            int f = 0;
            for (int j = 0; j < BB; ++j) f |= s_flag[j];
            s_fast = f ^ 1;   // fast == no batch crossed threshold
        }
        __syncthreads();

        const bool fast   = (s_fast != 0);
        const bool active = (t < el) && (n_tok < yl);
        const bool fire   = (!fast) && active && (accum_new >= 1.0f);
        const bool acc    = fast || ((!fast) && active && (accum_new < 1.0f));
        const float ak1 = 1.0f - accum;
        const float ak2 = alpha_j - ak1;

        const float4* ep = (const float4*)(eouts + ((size_t)b * TT + t) * DD + d0);
        const float4 e0 = ep[0], e1 = ep[1];
        const float e[8] = {e0.x, e0.y, e0.z, e0.w, e1.x, e1.y, e1.z, e1.w};

        if (fire) {
            float4 c0, c1;
            c0.x = st[0] + ak1 * e[0]; c0.y = st[1] + ak1 * e[1];
            c0.z = st[2] + ak1 * e[2]; c0.w = st[3] + ak1 * e[3];
            c1.x = st[4] + ak1 * e[4]; c1.y = st[5] + ak1 * e[5];
            c1.z = st[6] + ak1 * e[6]; c1.w = st[7] + ak1 * e[7];
            float* cvp = cv + ((size_t)b * YMAXX + n_tok) * DD + d0;
            ((float4*)cvp)[0] = c0;
            ((float4*)cvp)[1] = c1;
            #pragma unroll
            for (int i = 0; i < 8; ++i) st[i] = ak2 * e[i];
        } else if (acc) {
            #pragma unroll
            for (int i = 0; i < 8; ++i) st[i] += alpha_j * e[i];
        }

        if (lane == 0) {
            const float add1 = fire ? ak1 : (acc ? alpha_j : 0.0f);
            if ((fire || acc) && n_tok < YMAXX)
                aws[((size_t)b * YMAXX + n_tok) * TT + t] = add1;
            const int n_new = n_tok + (fire ? 1 : 0);
            if (fire && n_new < YMAXX)
                aws[((size_t)b * YMAXX + n_new) * TT + t] = ak2;
            s_ntok[b]  = n_new;
            s_accum[b] = fire ? ak2 : accum_new;
        }
        __syncthreads();
    }
}

// ---------------------------------------------------------------------------
extern "C" void kernel_launch(void* const* d_in, const int* in_sizes, int n_in,
                              void* d_out, int out_size, void* d_ws, size_t ws_size,
                              hipStream_t stream) {
    (void)in_sizes; (void)n_in; (void)ws_size;

    const float* eouts  = (const float*)d_in[0];
    const int*   elens  = (const int*)d_in[1];
    const int*   ylens  = (const int*)d_in[2];
    const float* conv_w = (const float*)d_in[3];
    const float* conv_b = (const float*)d_in[4];
    const float* ln_g   = (const float*)d_in[5];
    const float* ln_b   = (const float*)d_in[6];
    const float* proj_w = (const float*)d_in[7];
    const float* proj_b = (const float*)d_in[8];

    float* out    = (float*)d_out;
    float* cvp    = out + CV_OFF;
    float* awsp   = out + AWS_OFF;
    float* alphap = out + ALPHA_OFF;
    float* Bfrag  = (float*)d_ws;   // 1280*256 floats = 1.25 MB

    // cv/aws are sparsely written -> zero the output first (capture-legal)
    hipMemsetAsync(d_out, 0, (size_t)out_size * sizeof(float), stream);

    cif_pack_w<<<dim3(KK), dim3(DD), 0, stream>>>(conv_w, Bfrag);

    cif_alpha_kernel<<<dim3(TT / 16, BB), dim3(256), 0, stream>>>(
        eouts, Bfrag, conv_b, ln_g, ln_b, proj_w, proj_b, elens, alphap);

    cif_scan_kernel<<<dim3(1), dim3(512), 0, stream>>>(
        eouts, alphap, elens, ylens, cvp, awsp);
}